// Attentional_Aggregation_74990128988643
// MI455X (gfx1250) — compile-verified
//
#include <hip/hip_runtime.h>
#include <hip/hip_bf16.h>

typedef float v2f __attribute__((ext_vector_type(2)));
typedef float v8f __attribute__((ext_vector_type(8)));

#define DIM 128

// ---------------------------------------------------------------------------
// K1: last index of each contiguous (sorted) segment
// ---------------------------------------------------------------------------
__global__ __launch_bounds__(256) void k_lastidx(const int* __restrict__ seg,
                                                 int* __restrict__ last_idx, int n) {
    int i = blockIdx.x * blockDim.x + threadIdx.x;
    if (i >= n) return;
    int s = seg[i];
    if (i == n - 1 || seg[i + 1] != s) last_idx[s] = i;
}

// ---------------------------------------------------------------------------
// K2/K3: Out[r, c] = sum_k A[idx(r), k] * W[c, k] + bias[c]   (D = 128)
// Block = 256 threads (8 waves). Block computes 128 rows x 128 cols.
// Wave w owns the 16-row strip [blk*128 + 16w, +16), all 128 cols as 8 tiles.
// W^T staged in LDS packed as float2 (k, k+1) pairs per column: 64 KB.
// WMMA: V_WMMA_F32_16X16X4_F32, fp32-exact.
//   A 16x4 frag : lanes 0-15 hold M=0..15 {K=0,K=1}; lanes 16-31 {K=2,K=3}
//   B 4x16 frag : lanes 0-15 hold N=0..15 {K=0,K=1}; lanes 16-31 {K=2,K=3}
//   C/D 16x16   : VGPR v -> M=v (lanes 0-15) / M=v+8 (lanes 16-31), N=lane&15
// ---------------------------------------------------------------------------
__global__ __launch_bounds__(256, 2) void k_gemm_wmma(
    const float* __restrict__ A, const int* __restrict__ idx,
    const float* __restrict__ W, const float* __restrict__ bias,
    float* __restrict__ Out, int numRows) {

    __shared__ v2f s_wp[64 * 128];   // [k/2][col] -> (W[col][2k'], W[col][2k'+1]) : 64 KB

    const int tid = threadIdx.x;
    for (int t = tid; t < 64 * 128; t += 256) {
        int kk2 = t >> 7;
        int col = t & 127;
        const float* wp = W + (size_t)col * DIM + (kk2 << 1);
        v2f w;
        w.x = wp[0];
        w.y = wp[1];
        s_wp[t] = w;
    }
    __syncthreads();

    const int wave = tid >> 5;
    const int lane = tid & 31;
    const int m15  = lane & 15;
    const bool hi  = lane >= 16;

    const int rowstrip = blockIdx.x * 128 + wave * 16;

    // Per-lane source row for A-fragment loads (gather through idx if given)
    int arow_i = rowstrip + m15;
    if (arow_i >= numRows) arow_i = numRows - 1;   // clamp: garbage rows masked at store
    const int src = idx ? idx[arow_i] : arow_i;
    const float* arow = A + (size_t)src * DIM + (hi ? 2 : 0);

    // Accumulators seeded with the bias (same value down each column of C)
    v8f acc[8];
#pragma unroll
    for (int jt = 0; jt < 8; ++jt) {
        float b = bias[jt * 16 + m15];
        v8f c;
#pragma unroll
        for (int v = 0; v < 8; ++v) c[v] = b;
        acc[jt] = c;
    }

    const int koffhalf = hi ? 1 : 0;

#pragma unroll 4
    for (int k0 = 0; k0 < DIM; k0 += 4) {
        v2f a = *(const v2f*)(arow + k0);                 // global_load_b64
        int bidx = ((k0 >> 1) + koffhalf) * 128 + m15;    // LDS base for this K-chunk
#pragma unroll
        for (int jt = 0; jt < 8; ++jt) {
            v2f b = s_wp[bidx + jt * 16];                 // ds_load_b64
            acc[jt] = __builtin_amdgcn_wmma_f32_16x16x4_f32(
                /*neg_a=*/false, a, /*neg_b=*/false, b,
                /*c_mod=*/(short)0, acc[jt],
                /*reuse_a=*/false, /*reuse_b=*/false);
        }
    }

    // Epilogue: scatter D tiles (guard partial last block)
    const int rbase = rowstrip + (hi ? 8 : 0);
#pragma unroll
    for (int jt = 0; jt < 8; ++jt) {
        int col = jt * 16 + m15;
#pragma unroll
        for (int v = 0; v < 8; ++v) {
            int r = rbase + v;
            if (r < numRows) Out[(size_t)r * DIM + col] = acc[jt][v];
        }
    }
}

// ---------------------------------------------------------------------------
// K4: logits[i] = <q[seg[i]], keys[i]>   — one wave per row, float4 per lane
// ---------------------------------------------------------------------------
__global__ __launch_bounds__(256) void k_logits(const float* __restrict__ keys,
                                                const float* __restrict__ q,
                                                const int* __restrict__ seg,
                                                float* __restrict__ logits, int n) {
    int row  = (blockIdx.x * 256 + threadIdx.x) >> 5;
    int lane = threadIdx.x & 31;
    if (row >= n) return;
    int g = seg[row];
    float4 k4 = *(const float4*)(keys + (size_t)row * DIM + lane * 4);
    float4 q4 = *(const float4*)(q + (size_t)g * DIM + lane * 4);
    float p = k4.x * q4.x + k4.y * q4.y + k4.z * q4.z + k4.w * q4.w;
#pragma unroll
    for (int off = 16; off > 0; off >>= 1) p += __shfl_xor(p, off, 32);
    if (lane == 0) logits[row] = p;
}

// ---------------------------------------------------------------------------
// K5: fused per-group stable softmax + weighted key sum — one wave per group
// ---------------------------------------------------------------------------
__global__ __launch_bounds__(256) void k_group(const float* __restrict__ keys,
                                               const float* __restrict__ logits,
                                               const int* __restrict__ last_idx,
                                               float* __restrict__ out, int numG) {
    int g    = (blockIdx.x * 256 + threadIdx.x) >> 5;
    int lane = threadIdx.x & 31;
    if (g >= numG) return;

    int end   = last_idx[g];
    int start = (g == 0) ? 0 : last_idx[g - 1] + 1;

    // 1) segment max
    float m = -3.402823466e38f;
    for (int i = start + lane; i <= end; i += 32) m = fmaxf(m, logits[i]);
#pragma unroll
    for (int off = 16; off > 0; off >>= 1) m = fmaxf(m, __shfl_xor(m, off, 32));

    // 2) segment sum of exp
    float s = 0.0f;
    for (int i = start + lane; i <= end; i += 32) s += __expf(logits[i] - m);
#pragma unroll
    for (int off = 16; off > 0; off >>= 1) s += __shfl_xor(s, off, 32);
    float inv = 1.0f / s;

    // 3) weighted sum of keys: lane owns dims [4*lane, 4*lane+4)
    float4 acc = make_float4(0.f, 0.f, 0.f, 0.f);
    const float* kp = keys + lane * 4;
    for (int i = start; i <= end; ++i) {
        if (i + 2 <= end)
            __builtin_prefetch(kp + (size_t)(i + 2) * DIM, 0, 0);
        float w = __expf(logits[i] - m) * inv;
        float4 k4 = *(const float4*)(kp + (size_t)i * DIM);
        acc.x += w * k4.x;
        acc.y += w * k4.y;
        acc.z += w * k4.z;
        acc.w += w * k4.w;
    }
    *(float4*)(out + (size_t)g * DIM + lane * 4) = acc;
}

// ---------------------------------------------------------------------------
// Launch: last_idx -> q-GEMM (gathered) -> keys-GEMM -> logits -> group agg
// ---------------------------------------------------------------------------
extern "C" void kernel_launch(void* const* d_in, const int* in_sizes, int n_in,
                              void* d_out, int out_size, void* d_ws, size_t ws_size,
                              hipStream_t stream) {
    const float* emb = (const float*)d_in[0];
    const int*   seg = (const int*)d_in[1];
    const float* Wq  = (const float*)d_in[2];
    const float* bq  = (const float*)d_in[3];
    const float* Wk  = (const float*)d_in[4];
    const float* bk  = (const float*)d_in[5];
    float* out = (float*)d_out;

    const int N = in_sizes[1];          // number of elements
    const int G = out_size / DIM;       // number of groups

    // workspace layout: keys [N*D] | q [G*D] | logits [N] | last_idx [G]
    char*  ws       = (char*)d_ws;
    float* keys     = (float*)ws;
    float* q        = (float*)(ws + (size_t)N * DIM * sizeof(float));
    float* logits   = q + (size_t)G * DIM;
    int*   last_idx = (int*)(logits + N);

    k_lastidx<<<(N + 255) / 256, 256, 0, stream>>>(seg, last_idx, N);
    k_gemm_wmma<<<(G + 127) / 128, 256, 0, stream>>>(emb, last_idx, Wq, bq, q, G);
    k_gemm_wmma<<<(N + 127) / 128, 256, 0, stream>>>(emb, nullptr, Wk, bk, keys, N);
    k_logits<<<(N + 7) / 8, 256, 0, stream>>>(keys, q, seg, logits, N);
    k_group<<<(G + 7) / 8, 256, 0, stream>>>(keys, logits, last_idx, out, G);
}